// ProposalLayer_82411832475737
// MI455X (gfx1250) — compile-verified
//
#include <hip/hip_runtime.h>

// ------------------------------------------------------------------
// RPN ProposalLayer for MI455X (gfx1250, wave32)
// Pipeline: radix-select top-6000 (parallel suffix-scan select) ->
// LDS bitonic sort -> WMMA-assisted IoU bitmask -> single-wave
// software-pipelined greedy NMS -> [1,300,5] output.
// ------------------------------------------------------------------

#define NTOT   518400      // K*H*W candidates
#define HW     57600       // H*W
#define PPRE   6000        // pre-NMS count (NTOT > 6000 always)
#define NWORD  94          // ceil(6000/64) suppression words per row
#define NPOST  300
#define IOU_THR 0.6f

typedef unsigned int       u32;
typedef unsigned long long u64;
typedef __attribute__((ext_vector_type(2))) float v2f;
typedef __attribute__((ext_vector_type(8))) float v8f;

// ---- workspace byte offsets (ws base is 256B aligned from hipMalloc) ----
#define WS_HIST1   0          // u32[4096]   key bits [31:20]
#define WS_HIST2   16384      // u32[4096]   key bits [19:8]
#define WS_HIST3   32768      // u32[256]    key bits [7:0]
#define WS_PARAMS  36864      // u32: 0 b1,1 need1,2 b2,3 need2,4 b3,5 need3,6 T,7 cntA,8 cntB
#define WS_LISTA   40960      // u64[6144]   strictly-above-threshold (key,idx)
#define WS_LISTB   90112      // int[2048]   exactly-at-threshold indices
#define WS_SCORES  98304      // float[6016] sorted scores (+pad)
#define WS_BOXES   131072     // float[6016*4] sorted boxes (+pad)
#define WS_MASK    262144     // u64[6000*94] IoU>thr bitmask (j>i)
// total ~4.78 MB

// ---------------- helpers ----------------

// score for candidate i (k-major): cls_scores[0, 2k, h, w]
__device__ __forceinline__ float score_at(const float* __restrict__ cls, int i) {
  int k  = i / HW;
  int hw = i - k * HW;
  return cls[k * (2 * HW) + hw];
}

// monotonic key: larger float -> larger u32
__device__ __forceinline__ u32 fkey(float f) {
  u32 u = __float_as_uint(f);
  return (u & 0x80000000u) ? ~u : (u | 0x80000000u);
}

// box value for candidate row p, column c, replicating the torch view/permute
// reinterpretation of the [H,W,K,4] region buffer as [K,4,H,W].
__device__ __forceinline__ float box_at(const float* __restrict__ anch,
                                        const float* __restrict__ bdel,
                                        int p, int c) {
  int h2  = p / 2160;            // 2160 = W*K
  int rem = p - h2 * 2160;
  int w2  = rem / 9;
  int k2  = rem - w2 * 9;
  int f   = k2 * 230400 + c * 57600 + h2 * 240 + w2; // flat index into [H,W,K,4]
  // decode f back to original (h,w,k,cc)
  int h  = f / 8640;             // 8640 = W*K*4
  int r2 = f - h * 8640;
  int w  = r2 / 36;              // 36 = K*4
  int r3 = r2 - w * 36;
  int k  = r3 >> 2;
  int cc = r3 & 3;
  float v = anch[f] + bdel[(k * 4 + cc) * HW + h * 240 + w];
  return fminf(fmaxf(v, 0.0f), 1920.0f);
}

// ---------------- kernels ----------------

__global__ void init_zero_kernel(u32* __restrict__ wsu) {
  // zero bytes [0, 40960): hist1/2/3 + params.  <<<10,1024>>>
  wsu[blockIdx.x * 1024 + threadIdx.x] = 0u;
}

// mode 0: hist over bits[31:20]; mode 1: bits[19:8] within b1; mode 2: bits[7:0] within (b1,b2)
__global__ void hist_kernel(const float* __restrict__ cls, u32* __restrict__ wsu, int mode) {
  u32* hist1 = wsu;
  u32* hist2 = wsu + 4096;
  u32* hist3 = wsu + 8192;
  const u32* par = wsu + (WS_PARAMS / 4);
  int i = blockIdx.x * blockDim.x + threadIdx.x;
  if (i >= NTOT) return;
  if (mode == 0 && i + 2048 < NTOT) {
    int j = i + 2048;
    int k = j / HW;
    __builtin_prefetch(cls + (k * (2 * HW) + (j - k * HW)), 0, 0); // global_prefetch_b8
  }
  u32 u = fkey(score_at(cls, i));
  if (mode == 0) {
    atomicAdd(&hist1[u >> 20], 1u);
  } else if (mode == 1) {
    if ((u >> 20) == par[0]) atomicAdd(&hist2[(u >> 8) & 0xFFFu], 1u);
  } else {
    if ((u >> 8) == ((par[0] << 12) | par[2])) atomicAdd(&hist3[u & 0xFFu], 1u);
  }
}

// Parallel bin select: suffix-scan the histogram in LDS (Hillis-Steele,
// ping-pong buffers), then the unique thread whose bin satisfies
// S[b] >= need > S[b+1] records (bin, remaining-need).
__global__ __launch_bounds__(1024) void select_kernel(u32* __restrict__ wsu, int phase) {
  __shared__ u32 sa[4096];
  __shared__ u32 sb[4096];
  u32* par = wsu + (WS_PARAMS / 4);
  const u32* hist = (phase == 0) ? wsu : ((phase == 1) ? (wsu + 4096) : (wsu + 8192));
  const int nb    = (phase == 2) ? 256 : 4096;
  const u32 need  = (phase == 0) ? (u32)PPRE : ((phase == 1) ? par[1] : par[3]);
  int tid = threadIdx.x;

  for (int i = tid; i < 4096; i += 1024) sa[i] = (i < nb) ? hist[i] : 0u;
  __syncthreads();

  u32* src = sa;
  u32* dst = sb;
  for (u32 off = 1; off < 4096u; off <<= 1) {
    for (int i = tid; i < 4096; i += 1024)
      dst[i] = src[i] + ((i + (int)off < 4096) ? src[i + off] : 0u);
    __syncthreads();
    u32* t = src; src = dst; dst = t;
  }
  // src[b] = inclusive suffix sum S[b]
  for (int b = tid; b < nb; b += 1024) {
    u32 Sb  = src[b];
    u32 Sb1 = (b + 1 < 4096) ? src[b + 1] : 0u;
    if (Sb >= need && Sb1 < need) {           // exactly one thread matches
      u32 rem = need - Sb1;
      if (phase == 0)      { par[0] = (u32)b; par[1] = rem; }
      else if (phase == 1) { par[2] = (u32)b; par[3] = rem; }
      else {
        par[4] = (u32)b; par[5] = rem;
        par[6] = (par[0] << 20) | (par[2] << 8) | (u32)b; // threshold key T
        par[7] = 0u; par[8] = 0u;                         // reset compaction counters
      }
    }
  }
}

__global__ void compact_kernel(const float* __restrict__ cls, u32* __restrict__ wsu,
                               u64* __restrict__ listA, int* __restrict__ listB) {
  u32* par = wsu + (WS_PARAMS / 4);
  int i = blockIdx.x * blockDim.x + threadIdx.x;
  if (i >= NTOT) return;
  u32 u = fkey(score_at(cls, i));
  u32 T = par[6];
  if (u > T) {
    u32 pos = atomicAdd(&par[7], 1u);
    if (pos < 6144u) listA[pos] = (((u64)(~u)) << 32) | (u32)i; // asc sort => score desc, idx asc
  } else if (u == T) {
    u32 pos = atomicAdd(&par[8], 1u);
    if (pos < 2048u) listB[pos] = i;
  }
}

// One workgroup: bitonic-sort 8192 u64 keys in 64KB LDS, then materialize
// sorted scores + scrambled boxes for the top-6000.
__global__ __launch_bounds__(1024) void sort_build_kernel(
    const float* __restrict__ cls, const float* __restrict__ anch,
    const float* __restrict__ bdel, const u64* __restrict__ listA,
    const int* __restrict__ listB, const u32* __restrict__ wsu_params,
    float* __restrict__ scores, float* __restrict__ boxes) {
  __shared__ u64 s_keys[8192]; // 64 KB (CDNA5: 320 KB LDS per WGP)
  int tid = threadIdx.x;
  u32 cntA = wsu_params[7]; if (cntA > 6000u) cntA = 6000u;
  u32 T    = wsu_params[6];
  u32 cntB = wsu_params[8]; if (cntB > 2048u) cntB = 2048u;
  if (cntB > 8192u - cntA) cntB = 8192u - cntA;

  for (int i = tid; i < 8192; i += 1024) {
    u64 key;
    if ((u32)i < cntA)             key = listA[i];
    else if ((u32)i < cntA + cntB) key = (((u64)(~T)) << 32) | (u32)listB[i - cntA];
    else                           key = ~0ull;
    s_keys[i] = key;
  }
  __syncthreads();

  for (u32 k = 2; k <= 8192u; k <<= 1) {
    for (u32 j = k >> 1; j > 0; j >>= 1) {
      for (int i = tid; i < 8192; i += 1024) {
        int ixj = i ^ (int)j;
        if (ixj > i) {
          u64 a = s_keys[i], b = s_keys[ixj];
          bool up = ((((u32)i) & k) == 0);
          if (up ? (a > b) : (a < b)) { s_keys[i] = b; s_keys[ixj] = a; }
        }
      }
      __syncthreads();
    }
  }

  for (int p = tid; p < 6016; p += 1024) {
    if (p < PPRE) {
      int idx = (int)(u32)(s_keys[p] & 0xFFFFFFFFull);
      scores[p] = score_at(cls, idx);
#pragma unroll
      for (int c = 0; c < 4; ++c) boxes[p * 4 + c] = box_at(anch, bdel, idx, c);
    } else { // pad so the mask kernel's float4 j-loads stay clean
      scores[p] = 0.0f;
#pragma unroll
      for (int c = 0; c < 4; ++c) boxes[p * 4 + c] = 0.0f;
    }
  }
}

// One wave per 16(i) x 64(j) tile. The 16x16 area-sum matrix per sub-tile is
// computed with V_WMMA_F32_16X16X4_F32 (rank-2 outer sum: A=[area_i,1], B=[1,area_j]^T),
// whose C/D VGPR layout (lane L, reg v -> m=v+8*(L>=16), n=L&15) matches the
// per-lane IoU pair assignment and the ballot-based bit assembly.
__global__ __launch_bounds__(32) void mask_kernel(const float* __restrict__ boxes,
                                                  u64* __restrict__ mask) {
  __shared__ float s_bi[16 * 4];
  __shared__ float s_ai[16];
  __shared__ float s_bj[64 * 4];
  __shared__ float s_aj[64];
  int lane = threadIdx.x;
  int wj = blockIdx.x;           // 0..93  (j word)
  int ti = blockIdx.y;           // 0..374 (i tile)
  int i0 = ti * 16, j0 = wj * 64;

  if (lane < 16) {
    float4 b = ((const float4*)boxes)[i0 + lane];
    s_bi[lane * 4 + 0] = b.x; s_bi[lane * 4 + 1] = b.y;
    s_bi[lane * 4 + 2] = b.z; s_bi[lane * 4 + 3] = b.w;
    s_ai[lane] = fmaxf(b.z - b.x, 0.0f) * fmaxf(b.w - b.y, 0.0f);
  }
#pragma unroll
  for (int q = 0; q < 2; ++q) {
    int n = lane * 2 + q;                       // j0+n <= 6015 < 6016 (padded)
    float4 b = ((const float4*)boxes)[j0 + n];
    s_bj[n * 4 + 0] = b.x; s_bj[n * 4 + 1] = b.y;
    s_bj[n * 4 + 2] = b.z; s_bj[n * 4 + 3] = b.w;
    s_aj[n] = fmaxf(b.z - b.x, 0.0f) * fmaxf(b.w - b.y, 0.0f);
  }
  __syncthreads();

  bool lo = lane < 16;
  // A (16x4 f32): VGPR0 = K0 (lanes 0-15) / K2 (lanes 16-31); VGPR1 = K1 / K3
  v2f a;
  a[0] = lo ? s_ai[lane] : 0.0f;  // col K=0 = area_i
  a[1] = lo ? 1.0f : 0.0f;        // col K=1 = 1

  u64 myword = 0;
#pragma unroll
  for (int t = 0; t < 4; ++t) {
    // B (4x16 f32): VGPR0 = row K0 / row K2; VGPR1 = row K1 / row K3
    v2f b;
    b[0] = lo ? 1.0f : 0.0f;                  // row K=0 = 1
    b[1] = lo ? s_aj[t * 16 + lane] : 0.0f;   // row K=1 = area_j
    v8f cacc = {};
    v8f d = __builtin_amdgcn_wmma_f32_16x16x4_f32(
        false, a, false, b, (short)0, cacc, false, false);
    // d[v] = area_i[i0+m] + area_j[j0+t*16+n] for this lane's (m,n)

    int n  = t * 16 + (lane & 15);
    int gj = j0 + n;
    float jx1 = s_bj[n * 4 + 0], jy1 = s_bj[n * 4 + 1];
    float jx2 = s_bj[n * 4 + 2], jy2 = s_bj[n * 4 + 3];
#pragma unroll
    for (int v = 0; v < 8; ++v) {
      int m  = v + ((lane >> 4) << 3);
      int gi = i0 + m;
      float ix1 = s_bi[m * 4 + 0], iy1 = s_bi[m * 4 + 1];
      float ix2 = s_bi[m * 4 + 2], iy2 = s_bi[m * 4 + 3];
      float ww = fmaxf(fminf(ix2, jx2) - fmaxf(ix1, jx1), 0.0f);
      float hh = fmaxf(fminf(iy2, jy2) - fmaxf(iy1, jy1), 0.0f);
      float inter = ww * hh;
      float iou = inter / fmaxf(d[v] - inter, 1e-9f);
      bool pred = (gj > gi) && (gj < PPRE) && (iou > IOU_THR);
      u32 b32 = (u32)__ballot(pred);   // wave32: low 16 = row v, high 16 = row v+8
      if (lane == v)     myword |= ((u64)(b32 & 0xFFFFu)) << (t * 16);
      if (lane == v + 8) myword |= ((u64)(b32 >> 16)) << (t * 16);
    }
  }
  if (lane < 16) mask[(size_t)(i0 + lane) * NWORD + wj] = myword;
}

__device__ __forceinline__ u64 shfl_u64(u64 x, int src) {
  int lo = __shfl((int)(u32)x, src, 32);
  int hi = __shfl((int)(x >> 32), src, 32);
  return (((u64)(u32)hi) << 32) | (u32)lo;
}

// Single wave32: greedy sequential NMS over the bitmask. The 94-word
// suppression vector lives distributed across lanes (3 u64 each) so the
// 6000-step dependent loop needs only lane shuffles. Row i+1 is prefetched
// unconditionally (4.5MB total -- free at 23.3 TB/s) so the critical path
// per iteration is shuffle+OR, not a dependent global load.
__global__ __launch_bounds__(32) void nms_out_kernel(const u64* __restrict__ mask,
                                                     const float* __restrict__ scores,
                                                     const float* __restrict__ boxes,
                                                     float* __restrict__ out) {
  __shared__ int s_keep[NPOST];
  int lane = threadIdx.x;
  for (int r = lane; r < NPOST; r += 32) s_keep[r] = 0; // nonzero fill_value=0
  __syncthreads();

  u64 r0 = 0, r1 = 0, r2 = 0; // lane owns suppression words lane, lane+32, lane+64
  // prefetch row 0
  u64 n0 = mask[lane];
  u64 n1 = (lane + 32 < NWORD) ? mask[lane + 32] : 0ull;
  u64 n2 = (lane + 64 < NWORD) ? mask[lane + 64] : 0ull;
  int kept = 0;
  for (int i = 0; i < PPRE; ++i) {
    u64 c0 = n0, c1 = n1, c2 = n2;
    if (i + 1 < PPRE) {                      // prefetch next row
      const u64* nr = mask + (size_t)(i + 1) * NWORD;
      n0 = nr[lane];
      n1 = (lane + 32 < NWORD) ? nr[lane + 32] : 0ull;
      n2 = (lane + 64 < NWORD) ? nr[lane + 64] : 0ull;
    }
    int w = i >> 6;
    int slot = w >> 5, owner = w & 31;
    u64 val = (slot == 0) ? r0 : ((slot == 1) ? r1 : r2);
    u64 v = shfl_u64(val, owner);               // broadcast owner's word (uniform)
    if (!((v >> (i & 63)) & 1ull)) {
      if (lane == 0) s_keep[kept] = i;
      kept++;
      if (kept >= NPOST) break;                 // later keeps don't affect output
      r0 |= c0; r1 |= c1; r2 |= c2;
    }
  }
  __syncthreads();

  for (int r = lane; r < NPOST; r += 32) {
    int kidx = s_keep[r];
    out[r * 5 + 0] = scores[kidx];
    out[r * 5 + 1] = boxes[kidx * 4 + 0];
    out[r * 5 + 2] = boxes[kidx * 4 + 1];
    out[r * 5 + 3] = boxes[kidx * 4 + 2];
    out[r * 5 + 4] = boxes[kidx * 4 + 3];
  }
}

// ---------------- launch ----------------

extern "C" void kernel_launch(void* const* d_in, const int* in_sizes, int n_in,
                              void* d_out, int out_size, void* d_ws, size_t ws_size,
                              hipStream_t stream) {
  (void)in_sizes; (void)n_in; (void)out_size; (void)ws_size;
  const float* cls  = (const float*)d_in[0]; // [1, 2K, H, W]
  const float* bdel = (const float*)d_in[1]; // [1, 4K, H, W]
  const float* anch = (const float*)d_in[2]; // [H, W, K, 4]
  float* out = (float*)d_out;                // [1, 300, 5]
  char* ws = (char*)d_ws;

  u32*   wsu    = (u32*)ws;
  u32*   par    = (u32*)(ws + WS_PARAMS);
  u64*   listA  = (u64*)(ws + WS_LISTA);
  int*   listB  = (int*)(ws + WS_LISTB);
  float* scores = (float*)(ws + WS_SCORES);
  float* boxes  = (float*)(ws + WS_BOXES);
  u64*   mask   = (u64*)(ws + WS_MASK);

  const int EB = (NTOT + 255) / 256; // 2025

  init_zero_kernel<<<10, 1024, 0, stream>>>(wsu);

  hist_kernel<<<EB, 256, 0, stream>>>(cls, wsu, 0);
  select_kernel<<<1, 1024, 0, stream>>>(wsu, 0);
  hist_kernel<<<EB, 256, 0, stream>>>(cls, wsu, 1);
  select_kernel<<<1, 1024, 0, stream>>>(wsu, 1);
  hist_kernel<<<EB, 256, 0, stream>>>(cls, wsu, 2);
  select_kernel<<<1, 1024, 0, stream>>>(wsu, 2);

  compact_kernel<<<EB, 256, 0, stream>>>(cls, wsu, listA, listB);

  sort_build_kernel<<<1, 1024, 0, stream>>>(cls, anch, bdel, listA, listB, par,
                                            scores, boxes);

  mask_kernel<<<dim3(NWORD, PPRE / 16), 32, 0, stream>>>(boxes, mask);

  nms_out_kernel<<<1, 32, 0, stream>>>(mask, scores, boxes, out);
}